// PNALayer_27788438405447
// MI455X (gfx1250) — compile-verified
//
#include <hip/hip_runtime.h>
#include <hip/hip_bf16.h>
#include <math.h>

typedef float v2f __attribute__((ext_vector_type(2)));
typedef float v8f __attribute__((ext_vector_type(8)));

#define EPS_F 1e-5f
#define DIM 64
#define EDIM 16
#define KPRE 144    // 2*64 + 16
#define KPOST 832   // 13*64

__device__ __forceinline__ v8f wmma_f32(v2f a, v2f b, v8f c) {
    // V_WMMA_F32_16X16X4_F32: D = A(16x4) * B(4x16) + C(16x16), all fp32
    return __builtin_amdgcn_wmma_f32_16x16x4_f32(
        /*neg_a=*/false, a, /*neg_b=*/false, b,
        /*c_mod=*/(short)0, c, /*reuse_a=*/false, /*reuse_b=*/false);
}

__device__ __forceinline__ v2f ld2(const float* __restrict__ p) {
    const float2 t = *(const float2*)p;     // 8B-aligned: callers pass even offsets
    v2f r = { t.x, t.y };
    return r;
}

// ---------------------------------------------------------------- utilities
__global__ void fill_u32(unsigned* __restrict__ p, long n, unsigned v) {
    long i = (long)blockIdx.x * blockDim.x + threadIdx.x;
    long stride = (long)gridDim.x * blockDim.x;
    for (; i < n; i += stride) p[i] = v;
}

// -------------------------------------------- edge pre-MLP + atomic segment agg
// 1 wave (32 lanes) handles 16 edges; K-outer loop with 4 live accumulators so
// the gathered A-tile (node_feat[src]/[dst]/edge_feat) is loaded exactly once.
__global__ __launch_bounds__(256) void edge_pretrans_agg(
    const float* __restrict__ node_feat, const float* __restrict__ edge_feat,
    const float* __restrict__ pre_W, const float* __restrict__ pre_b,
    const int* __restrict__ src, const int* __restrict__ dst,
    float* __restrict__ deg, float* __restrict__ sum, float* __restrict__ sumsq,
    int* __restrict__ mx, int* __restrict__ mn)
{
    __shared__ float sW[KPRE * DIM];   // 36 KB
    __shared__ float sB[DIM];
    for (int i = threadIdx.x; i < KPRE * DIM; i += 256) sW[i] = pre_W[i];
    if (threadIdx.x < DIM) sB[threadIdx.x] = pre_b[threadIdx.x];
    __syncthreads();

    const int lane = threadIdx.x & 31;
    const int wid  = threadIdx.x >> 5;
    const int rlo  = lane & 15;
    const int hi   = lane >> 4;
    const long e0  = ((long)blockIdx.x * 8 + wid) * 16;

    // degree (one count per edge)
    if (lane < 16) {
        atomicAdd(&deg[dst[e0 + lane]], 1.0f);
    }

    const long e     = e0 + rlo;
    const long sbase = (long)src[e] * DIM;
    const long dbase = (long)dst[e] * DIM;

    v8f c[4];
    #pragma unroll
    for (int nt = 0; nt < 4; ++nt) {
        const float bias = sB[nt * 16 + rlo];
        #pragma unroll
        for (int v = 0; v < 8; ++v) c[nt][v] = bias;
    }

    #pragma unroll
    for (int k0 = 0; k0 < KPRE; k0 += 4) {
        const int kb = k0 + hi * 2;          // even
        v2f a;
        if (kb < 64)       a = ld2(node_feat + sbase + kb);
        else if (kb < 128) a = ld2(node_feat + dbase + (kb - 64));
        else               a = ld2(edge_feat + e * EDIM + (kb - 128));
        #pragma unroll
        for (int nt = 0; nt < 4; ++nt) {
            const int n = nt * 16 + rlo;
            v2f b = { sW[kb * DIM + n], sW[(kb + 1) * DIM + n] };
            c[nt] = wmma_f32(a, b, c[nt]);
        }
    }

    #pragma unroll
    for (int v = 0; v < 8; ++v) {
        const long base = (long)dst[e0 + v + hi * 8] * DIM;
        #pragma unroll
        for (int nt = 0; nt < 4; ++nt) {
            const float val = fmaxf(c[nt][v], 0.0f);
            const long  o   = base + nt * 16 + rlo;
            atomicAdd(&sum[o], val);
            atomicAdd(&sumsq[o], val * val);
            atomicMax(&mx[o], __float_as_int(val));   // val >= 0: int order == float order
            atomicMin(&mn[o], __float_as_int(val));
        }
    }
}

// ------------------------------------------------------- per-node scaler params
__global__ void node_params(const float* __restrict__ deg,
                            float4* __restrict__ nparams, int N)
{
    int i = blockIdx.x * blockDim.x + threadIdx.x;
    if (i >= N) return;
    float d    = deg[i];
    float degc = fmaxf(d, 1.0f);
    float inv  = 1.0f / degc;
    float hasf = (d > 0.0f) ? 1.0f : 0.0f;
    float logD = logf(d + 1.0f);
    float ampS = logD;                                        // AVG_LOG == 1.0
    float attS = (d > 0.0f) ? (1.0f / fmaxf(logD, EPS_F)) : 0.0f;
    nparams[i] = make_float4(inv, ampS, attS, hasf);
}

// -------------------------------------- paired h elements (k even, same 64-block)
__device__ __forceinline__ v2f h_elem2(
    const float* __restrict__ node_feat, const float* __restrict__ sum,
    const float* __restrict__ sumsq, const int* __restrict__ mx,
    const int* __restrict__ mn,
    float inv, float ampS, float attS, float hasf, long row, int k)
{
    if (k < 64) return ld2(node_feat + row * DIM + k);
    const int  j    = k - 64;
    const int  blk  = j >> 6;        // 0..11
    const int  cix  = j & 63;        // even
    const int  base = blk & 3;       // mean / max / min / std
    const int  sc   = blk >> 2;      // identity / amp / att
    const long o    = row * DIM + cix;
    const v2f  s2   = ld2(sum + o);
    const float m0 = s2[0] * inv, m1 = s2[1] * inv;
    v2f val;
    if (base == 0) { val[0] = m0; val[1] = m1; }
    else if (base == 1) {
        const int2 t = *(const int2*)(mx + o);
        val[0] = (hasf > 0.0f) ? __int_as_float(t.x) : 0.0f;
        val[1] = (hasf > 0.0f) ? __int_as_float(t.y) : 0.0f;
    } else if (base == 2) {
        const int2 t = *(const int2*)(mn + o);
        val[0] = (hasf > 0.0f) ? __int_as_float(t.x) : 0.0f;
        val[1] = (hasf > 0.0f) ? __int_as_float(t.y) : 0.0f;
    } else {
        const v2f q2 = ld2(sumsq + o);
        const float v0 = fmaxf(q2[0] * inv - m0 * m0, 0.0f);
        const float v1 = fmaxf(q2[1] * inv - m1 * m1, 0.0f);
        val[0] = (hasf > 0.0f) ? sqrtf(v0 + EPS_F) : 0.0f;
        val[1] = (hasf > 0.0f) ? sqrtf(v1 + EPS_F) : 0.0f;
    }
    if (sc == 1)      { val[0] *= ampS; val[1] *= ampS; }
    else if (sc == 2) { val[0] *= attS; val[1] *= attS; }
    return val;
}

// ------------------------------------- GEMM1 [N,832]x[832,64] + BN partial stats
// 1 wave = 16 nodes, 2 waves / block; K-outer with 4 accumulators so the
// expensive h_elem generation runs once per K-step instead of 4x.
__global__ __launch_bounds__(64) void gemm1_bnstats(
    const float* __restrict__ node_feat, const float* __restrict__ sum,
    const float* __restrict__ sumsq, const int* __restrict__ mx,
    const int* __restrict__ mn, const float4* __restrict__ nparams,
    const float* __restrict__ W1, const float* __restrict__ b1,
    float* __restrict__ x1, float* __restrict__ bnsum, float* __restrict__ bnsq)
{
    const int lane = threadIdx.x & 31;
    const int wid  = threadIdx.x >> 5;
    const int rlo  = lane & 15;
    const int hi   = lane >> 4;
    const long r0  = ((long)blockIdx.x * 2 + wid) * 16;
    const long row = r0 + rlo;

    const float4 p = nparams[row];
    const float inv = p.x, ampS = p.y, attS = p.z, hasf = p.w;

    v8f c[4];
    #pragma unroll
    for (int nt = 0; nt < 4; ++nt) {
        const float bias = b1[nt * 16 + rlo];
        #pragma unroll
        for (int v = 0; v < 8; ++v) c[nt][v] = bias;
    }

    for (int k0 = 0; k0 < KPOST; k0 += 4) {
        const int kb = k0 + hi * 2;          // even, same 64-block as kb+1
        v2f a = h_elem2(node_feat, sum, sumsq, mx, mn, inv, ampS, attS, hasf, row, kb);
        #pragma unroll
        for (int nt = 0; nt < 4; ++nt) {
            const int n = nt * 16 + rlo;
            v2f b = { W1[kb * DIM + n], W1[(kb + 1) * DIM + n] };
            c[nt] = wmma_f32(a, b, c[nt]);
        }
    }

    #pragma unroll
    for (int nt = 0; nt < 4; ++nt) {
        const int n = nt * 16 + rlo;
        float s = 0.0f, sq = 0.0f;
        #pragma unroll
        for (int v = 0; v < 8; ++v) {
            const float val = fmaxf(c[nt][v], 0.0f);
            x1[(r0 + v + hi * 8) * DIM + n] = val;
            s  += val;
            sq += val * val;
        }
        atomicAdd(&bnsum[n], s);
        atomicAdd(&bnsq[n], sq);
    }
}

// ----------------------------------------------------------- BN finalize (folded)
__global__ void bn_finalize(const float* __restrict__ bnsum, const float* __restrict__ bnsq,
                            const float* __restrict__ gamma, const float* __restrict__ beta,
                            float* __restrict__ bnscale, float* __restrict__ bnshift, float invN)
{
    int i = threadIdx.x;
    if (i < DIM) {
        float mu   = bnsum[i] * invN;
        float var  = bnsq[i] * invN - mu * mu;
        float rstd = rsqrtf(var + EPS_F);
        float scl  = gamma[i] * rstd;
        bnscale[i] = scl;
        bnshift[i] = beta[i] - scl * mu;
    }
}

// --------------------------------- GEMM2 [N,64]x[64,64] + relu + residual
__global__ __launch_bounds__(64) void gemm2_out(
    const float* __restrict__ x1, const float* __restrict__ bnscale,
    const float* __restrict__ bnshift, const float* __restrict__ W2,
    const float* __restrict__ b2, const float* __restrict__ node_feat,
    float* __restrict__ out)
{
    __shared__ float sW[DIM * DIM];   // 16 KB
    __shared__ float sScale[DIM], sShift[DIM], sB2[DIM];
    for (int i = threadIdx.x; i < DIM * DIM; i += 64) sW[i] = W2[i];
    if (threadIdx.x < DIM) {
        sScale[threadIdx.x] = bnscale[threadIdx.x];
        sShift[threadIdx.x] = bnshift[threadIdx.x];
        sB2[threadIdx.x]    = b2[threadIdx.x];
    }
    __syncthreads();

    const int lane = threadIdx.x & 31;
    const int wid  = threadIdx.x >> 5;
    const int rlo  = lane & 15;
    const int hi   = lane >> 4;
    const long r0  = ((long)blockIdx.x * 2 + wid) * 16;
    const long row = r0 + rlo;

    v8f c[4];
    #pragma unroll
    for (int nt = 0; nt < 4; ++nt) {
        const float bias = sB2[nt * 16 + rlo];
        #pragma unroll
        for (int v = 0; v < 8; ++v) c[nt][v] = bias;
    }

    #pragma unroll
    for (int k0 = 0; k0 < DIM; k0 += 4) {
        const int kb = k0 + hi * 2;          // even
        const v2f x = ld2(x1 + row * DIM + kb);
        v2f a = { x[0] * sScale[kb]     + sShift[kb],
                  x[1] * sScale[kb + 1] + sShift[kb + 1] };
        #pragma unroll
        for (int nt = 0; nt < 4; ++nt) {
            const int n = nt * 16 + rlo;
            v2f b = { sW[kb * DIM + n], sW[(kb + 1) * DIM + n] };
            c[nt] = wmma_f32(a, b, c[nt]);
        }
    }

    #pragma unroll
    for (int v = 0; v < 8; ++v) {
        const long orow = r0 + v + hi * 8;
        #pragma unroll
        for (int nt = 0; nt < 4; ++nt) {
            const int n = nt * 16 + rlo;
            out[orow * DIM + n] = fmaxf(c[nt][v], 0.0f) + node_feat[orow * DIM + n];
        }
    }
}

// ---------------------------------------------------------------------- launch
extern "C" void kernel_launch(void* const* d_in, const int* in_sizes, int n_in,
                              void* d_out, int out_size, void* d_ws, size_t ws_size,
                              hipStream_t stream) {
    const float* node_feat = (const float*)d_in[0];
    const float* edge_feat = (const float*)d_in[1];
    const float* pre_W     = (const float*)d_in[2];
    const float* pre_b     = (const float*)d_in[3];
    const float* W1        = (const float*)d_in[4];
    const float* b1        = (const float*)d_in[5];
    const float* gamma     = (const float*)d_in[6];
    const float* beta      = (const float*)d_in[7];
    const float* W2        = (const float*)d_in[8];
    const float* b2        = (const float*)d_in[9];
    const int*   src       = (const int*)d_in[10];
    const int*   dst       = (const int*)d_in[11];

    const long N = in_sizes[0] / DIM;       // 100000
    const long E = in_sizes[10];            // 1600000

    // workspace layout (floats), ~130 MB total
    float*  deg     = (float*)d_ws;                 // N
    float*  sum     = deg + N;                      // N*64
    float*  sumsq   = sum + N * DIM;                // N*64
    int*    mx      = (int*)(sumsq + N * DIM);      // N*64
    int*    mn      = mx + N * DIM;                 // N*64
    float4* nparams = (float4*)(mn + N * DIM);      // N
    float*  x1      = (float*)(nparams + N);        // N*64
    float*  bnsum   = x1 + N * DIM;                 // 64
    float*  bnsq    = bnsum + DIM;                  // 64
    float*  bnscale = bnsq + DIM;                   // 64
    float*  bnshift = bnscale + DIM;                // 64

    // init accumulators (sum & sumsq contiguous -> one fill)
    fill_u32<<<1024, 256, 0, stream>>>((unsigned*)deg, N, 0u);
    fill_u32<<<2048, 256, 0, stream>>>((unsigned*)sum, N * DIM * 2, 0u);
    fill_u32<<<2048, 256, 0, stream>>>((unsigned*)mx, N * DIM, 0u);            // e>=0
    fill_u32<<<2048, 256, 0, stream>>>((unsigned*)mn, N * DIM, 0x7F800000u);   // +inf
    fill_u32<<<1, 128, 0, stream>>>((unsigned*)bnsum, 2 * DIM, 0u);

    edge_pretrans_agg<<<(int)(E / 128), 256, 0, stream>>>(
        node_feat, edge_feat, pre_W, pre_b, src, dst, deg, sum, sumsq, mx, mn);

    node_params<<<(int)((N + 255) / 256), 256, 0, stream>>>(deg, nparams, (int)N);

    gemm1_bnstats<<<(int)(N / 32), 64, 0, stream>>>(
        node_feat, sum, sumsq, mx, mn, nparams, W1, b1, x1, bnsum, bnsq);

    bn_finalize<<<1, 64, 0, stream>>>(bnsum, bnsq, gamma, beta, bnscale, bnshift,
                                      1.0f / (float)N);

    gemm2_out<<<(int)(N / 32), 64, 0, stream>>>(
        x1, bnscale, bnshift, W2, b2, node_feat, (float*)d_out);
}